// Decoder_43508018708769
// MI455X (gfx1250) — compile-verified
//
#include <hip/hip_runtime.h>
#include <hip/hip_bf16.h>
#include <math.h>

#define BB    64
#define UNITS 512
#define EMB   256
#define NS    32
#define NW    50
#define VOCAB 50000
#define OOV   100
#define EXTV  (VOCAB + OOV)   // 50100

typedef __attribute__((ext_vector_type(2))) float v2f;
typedef __attribute__((ext_vector_type(8))) float v8f;

// ---------------------------------------------------------------------------
// Kernel 1: GRU cell step (keras reset_after=True). One block per batch row.
// ---------------------------------------------------------------------------
__global__ __launch_bounds__(256) void k_gru(const float* __restrict__ x,
                                             const float* __restrict__ h0,
                                             const float* __restrict__ Wx,
                                             const float* __restrict__ Wh,
                                             const float* __restrict__ gb,
                                             float* __restrict__ dec) {
  __shared__ float sx[EMB];
  __shared__ float sh[UNITS];
  const int b = blockIdx.x, t = threadIdx.x;
  sx[t] = x[b * EMB + t];
  sh[t] = h0[b * UNITS + t];
  sh[t + 256] = h0[b * UNITS + t + 256];
  __syncthreads();
  for (int u = t; u < UNITS; u += 256) {
    float xg0 = gb[u], xg1 = gb[UNITS + u], xg2 = gb[2 * UNITS + u];
    float hg0 = gb[3 * UNITS + u], hg1 = gb[4 * UNITS + u], hg2 = gb[5 * UNITS + u];
    for (int k = 0; k < EMB; ++k) {
      const float xv = sx[k];
      const float* wr = Wx + k * (3 * UNITS) + u;
      xg0 += xv * wr[0];
      xg1 += xv * wr[UNITS];
      xg2 += xv * wr[2 * UNITS];
    }
    for (int k = 0; k < UNITS; ++k) {
      const float hv = sh[k];
      const float* wr = Wh + k * (3 * UNITS) + u;
      hg0 += hv * wr[0];
      hg1 += hv * wr[UNITS];
      hg2 += hv * wr[2 * UNITS];
    }
    const float z = 1.0f / (1.0f + expf(-(xg0 + hg0)));
    const float r = 1.0f / (1.0f + expf(-(xg1 + hg1)));
    const float hc = tanhf(xg2 + r * hg2);
    dec[b * UNITS + u] = z * sh[u] + (1.0f - z) * hc;
  }
}

// ---------------------------------------------------------------------------
// Kernel 2: sentence attention + context + hidden_t + qw + p_gen.
// One block per batch row (256 threads = 8 wave32).
// ---------------------------------------------------------------------------
__global__ __launch_bounds__(256) void k_attn(const float* __restrict__ dec,
                                              const float* __restrict__ encS,
                                              const float* __restrict__ Ws,
                                              const float* __restrict__ Ww,
                                              const float* __restrict__ Wd,
                                              const float* __restrict__ wc,
                                              const float* __restrict__ wh,
                                              const float* __restrict__ wi,
                                              const float* __restrict__ emb,
                                              float* __restrict__ qw_out,
                                              float* __restrict__ alpha_out,
                                              float* __restrict__ hid_out,
                                              float* __restrict__ pgen_out) {
  __shared__ float sdec[UNITS];
  __shared__ float sq[UNITS];
  __shared__ float sctx[UNITS];
  __shared__ float sred[256];
  __shared__ float ssc[NS];
  const int b = blockIdx.x, t = threadIdx.x;
  sdec[t] = dec[b * UNITS + t];
  sdec[t + 256] = dec[b * UNITS + t + 256];
  __syncthreads();

  // q = dec @ W_att_sent, qw = dec @ W_att_word (coalesced over columns)
  for (int u = t; u < UNITS; u += 256) {
    float aq = 0.0f, aw = 0.0f;
    for (int k = 0; k < UNITS; ++k) {
      const float dv = sdec[k];
      aq += dv * Ws[k * UNITS + u];
      aw += dv * Ww[k * UNITS + u];
    }
    sq[u] = aq;
    qw_out[b * UNITS + u] = aw;
  }
  __syncthreads();

  // score_s[s] = dot(q, encS[b,s,:])  (8 threads per sentence)
  {
    const int s = t >> 3, l8 = t & 7;
    const float* es = encS + ((size_t)b * NS + s) * UNITS;
    float p = 0.0f;
    for (int d = l8; d < UNITS; d += 8) p += sq[d] * es[d];
    sred[t] = p;
    __syncthreads();
    if (l8 == 0) {
      float v = 0.0f;
      for (int i = 0; i < 8; ++i) v += sred[t + i];
      ssc[s] = v;
    }
  }
  __syncthreads();

  // softmax over 32 sentences (first wave, wave32 shuffles)
  if (t < 32) {
    const float v = ssc[t];
    float m = v;
    for (int off = 16; off; off >>= 1) m = fmaxf(m, __shfl_down(m, off));
    m = __shfl(m, 0);
    const float e = expf(v - m);
    float sum = e;
    for (int off = 16; off; off >>= 1) sum += __shfl_down(sum, off);
    sum = __shfl(sum, 0);
    const float a = e / sum;
    ssc[t] = a;
    alpha_out[b * NS + t] = a;
  }
  __syncthreads();

  // context[d] = sum_s alpha[s] * encS[b,s,d]
  for (int d = t; d < UNITS; d += 256) {
    float c = 0.0f;
    for (int s = 0; s < NS; ++s) c += ssc[s] * encS[((size_t)b * NS + s) * UNITS + d];
    sctx[d] = c;
  }
  __syncthreads();

  // hidden_t[j] = tanh([ctx|dec] @ weight_dec), 256 outputs = 1 per thread
  {
    float a = 0.0f;
    for (int k = 0; k < UNITS; ++k) a += sctx[k] * Wd[k * EMB + t];
    for (int k = 0; k < UNITS; ++k) a += sdec[k] * Wd[(UNITS + k) * EMB + t];
    hid_out[b * EMB + t] = tanhf(a);
  }

  // p_gen = sigmoid(ctx.wc + dec.wh + emb.wi)
  {
    float p = 0.0f;
    for (int i = t; i < UNITS; i += 256) p += sctx[i] * wc[i] + sdec[i] * wh[i];
    p += emb[b * EMB + t] * wi[t];
    sred[t] = p;
    __syncthreads();
    for (int s = 128; s > 0; s >>= 1) {
      if (t < s) sred[t] += sred[t + s];
      __syncthreads();
    }
    if (t == 0) pgen_out[b] = 1.0f / (1.0f + expf(-sred[0]));
  }
}

// ---------------------------------------------------------------------------
// Kernel 3: word attention scores + softmax + beta. One block per (b,s).
// Dominant-bandwidth kernel: streams 210 MB of encoded_words once.
// ---------------------------------------------------------------------------
__global__ __launch_bounds__(256) void k_words(const float* __restrict__ qw,
                                               const float* __restrict__ alpha,
                                               const float* __restrict__ encW,
                                               float* __restrict__ beta_out) {
  __shared__ float sqw[UNITS];
  __shared__ float ssc[NW];
  __shared__ float snorm[2];
  const int bs = blockIdx.x;
  const int b = bs >> 5, s = bs & 31;
  const int t = threadIdx.x, lane = t & 31, wid = t >> 5;
  sqw[t] = qw[b * UNITS + t];
  sqw[t + 256] = qw[b * UNITS + t + 256];
  __syncthreads();

  const float* base = encW + (size_t)bs * NW * UNITS;
  for (int w = wid; w < NW; w += 8) {
    const float* ew = base + (size_t)w * UNITS;
    float p = 0.0f;
    for (int d = lane; d < UNITS; d += 32) p += sqw[d] * ew[d];
    for (int off = 16; off; off >>= 1) p += __shfl_down(p, off);
    if (lane == 0) ssc[w] = p;
  }
  __syncthreads();
  if (t == 0) {
    float m = ssc[0];
    for (int w = 1; w < NW; ++w) m = fmaxf(m, ssc[w]);
    float sum = 0.0f;
    for (int w = 0; w < NW; ++w) sum += expf(ssc[w] - m);
    snorm[0] = m;
    snorm[1] = alpha[b * NS + s] / sum;
  }
  __syncthreads();
  if (t < NW) beta_out[(size_t)bs * NW + t] = expf(ssc[t] - snorm[0]) * snorm[1];
}

// ---------------------------------------------------------------------------
// Kernel 4: deterministic scatter-add into copy_distr via 200KB LDS row
// accumulator (CDNA5: 320KB LDS/WGP). Slot ownership by (idx % 256) — no
// atomics, bitwise-deterministic. One block per batch row.
// ---------------------------------------------------------------------------
__global__ __launch_bounds__(256) void k_scatter(const float* __restrict__ beta,
                                                 const int* __restrict__ nidx,
                                                 float* __restrict__ cpy) {
  __shared__ float sacc[EXTV];       // 200,400 B
  __shared__ int   sidx[NS * NW];
  __shared__ float sb[NS * NW];
  const int b = blockIdx.x, t = threadIdx.x;
  for (int i = t; i < EXTV; i += 256) sacc[i] = 0.0f;
  for (int i = t; i < NS * NW; i += 256) {
    sidx[i] = nidx[b * NS * NW + i];
    sb[i] = beta[(size_t)b * NS * NW + i];
  }
  __syncthreads();
  for (int j = 0; j < NS * NW; ++j) {
    const int idx = sidx[j];
    if ((idx & 255) == t) sacc[idx] += sb[j];
  }
  __syncthreads();
  for (int i = t; i < EXTV; i += 256) cpy[(size_t)b * EXTV + i] = sacc[i];
}

// ---------------------------------------------------------------------------
// Kernel 5: fc logits GEMM via V_WMMA_F32_16X16X4_F32.
// M=64 (4 waves x 16 rows), N=50000 (3125 blocks x 16 cols), K=256 (64 WMMAs).
// Writes logits (pre-softmax) in-place into the gen_distr output region.
// ---------------------------------------------------------------------------
__global__ __launch_bounds__(128) void k_fc_wmma(const float* __restrict__ hid,
                                                 const float* __restrict__ W,
                                                 const float* __restrict__ bias,
                                                 float* __restrict__ gen) {
  const int lane = threadIdx.x & 31;
  const int wave = threadIdx.x >> 5;
  const int l16  = lane & 15;
  const int hi   = lane >> 4;       // 0 or 1
  const int hi2  = hi << 1;         // K sub-offset 0 or 2
  const int m0   = wave * 16;
  const int ncol = blockIdx.x * 16;

  v8f acc = {};
  const float* Arow = hid + (m0 + l16) * EMB;        // A: 16x4 fp32 fragment rows
  const float* Bcol = W + ncol + l16;                 // B: 4x16 fp32 fragment cols
#pragma unroll 4
  for (int k = 0; k < EMB; k += 4) {
    v2f a, bf;
    a.x  = Arow[k + hi2];
    a.y  = Arow[k + hi2 + 1];
    bf.x = Bcol[(size_t)(k + hi2) * VOCAB];
    bf.y = Bcol[(size_t)(k + hi2 + 1) * VOCAB];
    acc = __builtin_amdgcn_wmma_f32_16x16x4_f32(false, a, false, bf,
                                                (short)0, acc, false, false);
  }
  const float bv = bias[ncol + l16];
  const int rbase = m0 + (hi << 3);
#pragma unroll
  for (int r = 0; r < 8; ++r)
    gen[(size_t)(rbase + r) * EXTV + ncol + l16] = acc[r] + bv;
}

// ---------------------------------------------------------------------------
// Kernel 6: per-row max and 1/sum(exp) over the 50000 logits.
// ---------------------------------------------------------------------------
__global__ __launch_bounds__(256) void k_rowred(const float* __restrict__ gen,
                                                float* __restrict__ rowmax,
                                                float* __restrict__ rowinv) {
  __shared__ float red[256];
  const int b = blockIdx.x, t = threadIdx.x;
  const float* row = gen + (size_t)b * EXTV;
  float m = -3.402823466e38f;
  for (int j = t; j < VOCAB; j += 256) m = fmaxf(m, row[j]);
  red[t] = m;
  __syncthreads();
  for (int s = 128; s > 0; s >>= 1) {
    if (t < s) red[t] = fmaxf(red[t], red[t + s]);
    __syncthreads();
  }
  m = red[0];
  __syncthreads();
  float sum = 0.0f;
  for (int j = t; j < VOCAB; j += 256) sum += expf(row[j] - m);
  red[t] = sum;
  __syncthreads();
  for (int s = 128; s > 0; s >>= 1) {
    if (t < s) red[t] += red[t + s];
    __syncthreads();
  }
  if (t == 0) {
    rowmax[b] = m;
    rowinv[b] = 1.0f / red[0];
  }
}

// ---------------------------------------------------------------------------
// Kernel 7: in-place softmax + zero-pad the OOV columns.
// ---------------------------------------------------------------------------
__global__ __launch_bounds__(256) void k_gensm(float* __restrict__ gen,
                                               const float* __restrict__ rowmax,
                                               const float* __restrict__ rowinv) {
  const int i = blockIdx.x * 256 + threadIdx.x;
  if (i >= BB * EXTV) return;
  const int b = i / EXTV;
  const int j = i - b * EXTV;
  gen[i] = (j < VOCAB) ? expf(gen[i] - rowmax[b]) * rowinv[b] : 0.0f;
}

// ---------------------------------------------------------------------------
extern "C" void kernel_launch(void* const* d_in, const int* in_sizes, int n_in,
                              void* d_out, int out_size, void* d_ws, size_t ws_size,
                              hipStream_t stream) {
  const float* emb  = (const float*)d_in[0];   // [64,1,256]
  const float* h0   = (const float*)d_in[1];   // [64,512]
  const float* encS = (const float*)d_in[2];   // [64,32,512]
  const float* encW = (const float*)d_in[3];   // [64,32,50,512]
  const int*   nidx = (const int*)d_in[4];     // [64,1600]
  // d_in[5] = batch_size (unused)
  const float* Wx  = (const float*)d_in[6];    // [256,1536]
  const float* Wh  = (const float*)d_in[7];    // [512,1536]
  const float* gb  = (const float*)d_in[8];    // [2,1536]
  const float* Ws  = (const float*)d_in[9];    // [512,512]
  const float* Ww  = (const float*)d_in[10];   // [512,512]
  const float* fcW = (const float*)d_in[11];   // [256,50000]
  const float* fcb = (const float*)d_in[12];   // [50000]
  const float* wc  = (const float*)d_in[13];   // [512,1]
  const float* wh  = (const float*)d_in[14];   // [512,1]
  const float* Wd  = (const float*)d_in[15];   // [1024,256]
  const float* wi  = (const float*)d_in[16];   // [256,1]

  float* out = (float*)d_out;
  float* dec = out;                         // [64,512]
  float* gen = out + BB * UNITS;            // [64,50100] (logits, then softmax)
  float* cpy = gen + BB * EXTV;             // [64,50100]
  float* pg  = cpy + BB * EXTV;             // [64,1]

  float* ws  = (float*)d_ws;
  float* hid = ws;                          // [64,256]
  float* qw  = ws + 16384;                  // [64,512]
  float* alp = ws + 49152;                  // [64,32]
  float* rmx = ws + 51200;                  // [64]
  float* rin = ws + 51264;                  // [64]
  float* bet = ws + 51328;                  // [64,1600]

  k_gru    <<<BB, 256, 0, stream>>>(emb, h0, Wx, Wh, gb, dec);
  k_attn   <<<BB, 256, 0, stream>>>(dec, encS, Ws, Ww, Wd, wc, wh, wi, emb,
                                    qw, alp, hid, pg);
  k_words  <<<BB * NS, 256, 0, stream>>>(qw, alp, encW, bet);
  k_scatter<<<BB, 256, 0, stream>>>(bet, nidx, cpy);
  k_fc_wmma<<<VOCAB / 16, 128, 0, stream>>>(hid, fcW, fcb, gen);
  k_rowred <<<BB, 256, 0, stream>>>(gen, rmx, rin);
  k_gensm  <<<(BB * EXTV + 255) / 256, 256, 0, stream>>>(gen, rmx, rin);
}